// GraphAttentionToy_6073083757190
// MI455X (gfx1250) — compile-verified
//
#include <hip/hip_runtime.h>
#include <math.h>

// Problem constants (match reference)
constexpr int N    = 2048;
constexpr int DIN  = 256;
constexpr int DK   = 64;
constexpr int DV   = 64;
constexpr int DR   = 64;
constexpr int PROJ = DK + DK + DV + DR;            // 256
constexpr float SCALE = 0.08838834764831845f;      // 1/sqrt(DK+DR) = 1/sqrt(128)

typedef __attribute__((ext_vector_type(2))) float v2f;   // A/B frag: 16x4 f32 = 2 VGPRs/lane
typedef __attribute__((ext_vector_type(8))) float v8f;   // C/D frag: 16x16 f32 = 8 VGPRs/lane

__device__ __forceinline__ v8f wmma_f32_16x16x4(v2f a, v2f b, v8f c) {
  // 8 args: (neg_a, A, neg_b, B, c_mod, C, reuse_a, reuse_b)
  return __builtin_amdgcn_wmma_f32_16x16x4_f32(false, a, false, b, (short)0, c,
                                               false, false);
}

// -------------------------------------------------------------------------
// Kernel 1: proj = H @ W^T   (2048x256) = (2048x256)x(256x256)
// One wave per 16x16 output tile; K loop in steps of 4 via V_WMMA_F32_16X16X4_F32.
// A layout: lane<16 -> M=lane, K={k,k+1}; lane>=16 -> M=lane-16, K={k+2,k+3}.
// B layout mirrors A with N instead of M. B[k_,n] = W[n, k_] (contiguous in k).
// -------------------------------------------------------------------------
__global__ void __launch_bounds__(32)
proj_kernel(const float* __restrict__ H, const float* __restrict__ W,
            float* __restrict__ proj) {
  const int lane = threadIdx.x;
  const int half = lane >> 4;          // 0 or 1
  const int l15  = lane & 15;
  const int m0 = blockIdx.y * 16;
  const int n0 = blockIdx.x * 16;

  const float* arow = H + (size_t)(m0 + l15) * DIN + 2 * half;
  const float* brow = W + (size_t)(n0 + l15) * DIN + 2 * half;

  v8f acc = {};
#pragma unroll 8
  for (int k = 0; k < DIN; k += 4) {
    v2f a = *(const v2f*)(arow + k);   // b64 load
    v2f b = *(const v2f*)(brow + k);   // b64 load
    acc = wmma_f32_16x16x4(a, b, acc);
  }
#pragma unroll
  for (int v = 0; v < 8; ++v) {
    const int m = m0 + v + 8 * half;   // C/D layout: VGPR v, lane half selects +8
    proj[(size_t)m * PROJ + n0 + l15] = acc[v];
  }
}

// -------------------------------------------------------------------------
// Kernel 2: S[m,j] = SCALE * (Q[m].K[j] + R[m].D[m,j])
// This kernel streams the 1 GiB D tensor -> it IS the roofline bottleneck.
// One wave per 16x16 tile of S. Q.K^T done with WMMA; the R.D batched dot is
// per-lane float4 FMA streaming, written directly into the matching C-layout
// accumulator element (lane = column, v = row).
// -------------------------------------------------------------------------
__global__ void __launch_bounds__(32)
scores_kernel(const float* __restrict__ proj, const float* __restrict__ D,
              float* __restrict__ S) {
  const int lane = threadIdx.x;
  const int half = lane >> 4;
  const int l15  = lane & 15;
  const int m0 = blockIdx.y * 16;
  const int j0 = blockIdx.x * 16;

  // ---- Q @ K^T tile (K-dim = 64 -> 16 wmma ops) ----
  const float* qrow = proj + (size_t)(m0 + l15) * PROJ + 0  + 2 * half; // Q cols [0,64)
  const float* krow = proj + (size_t)(j0 + l15) * PROJ + DK + 2 * half; // K cols [64,128)
  v8f acc = {};
#pragma unroll
  for (int k = 0; k < DK; k += 4) {
    v2f a = *(const v2f*)(qrow + k);
    v2f b = *(const v2f*)(krow + k);
    acc = wmma_f32_16x16x4(a, b, acc);
  }

  // ---- RD[m,j] = R[m] . D[m,j,:]  (streams D: 64 KB per tile) ----
  const int j = j0 + l15;
#pragma unroll
  for (int v = 0; v < 8; ++v) {
    const int m = m0 + v + 8 * half;
    const float4* dp = (const float4*)(D + ((size_t)m * N + j) * DR);
    const float4* rp = (const float4*)(proj + (size_t)m * PROJ + 3 * DK); // R cols [192,256)
    if (v < 7) {
      // gfx1250 global_prefetch_b8 on the next D row this lane will consume
      __builtin_prefetch(D + ((size_t)(m + 1) * N + j) * DR, 0, 1);
    }
    float dot = 0.0f;
#pragma unroll
    for (int r = 0; r < DR / 4; ++r) {
      const float4 d4 = dp[r];         // b128 load, 256B-aligned rows
      const float4 r4 = rp[r];
      dot = fmaf(d4.x, r4.x, dot);
      dot = fmaf(d4.y, r4.y, dot);
      dot = fmaf(d4.z, r4.z, dot);
      dot = fmaf(d4.w, r4.w, dot);
    }
    S[(size_t)m * N + j] = SCALE * (acc[v] + dot);
  }
}

// -------------------------------------------------------------------------
// Kernel 3: in-place row softmax on S (2048 x 2048). One block (256 thr) per row.
// S is 16 MB -> L2 resident (192 MB), so the extra pass is nearly free.
// -------------------------------------------------------------------------
__global__ void __launch_bounds__(256)
softmax_kernel(float* __restrict__ S) {
  __shared__ float red[256];
  const int t = threadIdx.x;
  float* Srow = S + (size_t)blockIdx.x * N;

  float vals[8];
  float vmax = -INFINITY;
#pragma unroll
  for (int k = 0; k < 8; ++k) {
    vals[k] = Srow[t + k * 256];
    vmax = fmaxf(vmax, vals[k]);
  }
  red[t] = vmax;
  __syncthreads();
  for (int s = 128; s > 0; s >>= 1) {
    if (t < s) red[t] = fmaxf(red[t], red[t + s]);
    __syncthreads();
  }
  vmax = red[0];
  __syncthreads();

  float vsum = 0.0f;
#pragma unroll
  for (int k = 0; k < 8; ++k) {
    vals[k] = __expf(vals[k] - vmax);
    vsum += vals[k];
  }
  red[t] = vsum;
  __syncthreads();
  for (int s = 128; s > 0; s >>= 1) {
    if (t < s) red[t] += red[t + s];
    __syncthreads();
  }
  const float inv = 1.0f / red[0];
#pragma unroll
  for (int k = 0; k < 8; ++k) {
    Srow[t + k * 256] = vals[k] * inv;
  }
}

// -------------------------------------------------------------------------
// Kernel 4: out = P @ V   (2048x64) = (2048x2048)x(2048x64)
// One wave per 16x16 output tile, K loop over 2048 in steps of 4 (512 wmma).
// V lives in proj cols [128,192). B loads are column-strided scalars.
// -------------------------------------------------------------------------
__global__ void __launch_bounds__(32)
out_kernel(const float* __restrict__ P, const float* __restrict__ proj,
           float* __restrict__ out) {
  const int lane = threadIdx.x;
  const int half = lane >> 4;
  const int l15  = lane & 15;
  const int m0 = blockIdx.y * 16;
  const int n0 = blockIdx.x * 16;

  const float* arow = P + (size_t)(m0 + l15) * N + 2 * half;
  const float* bcol = proj + 2 * DK + n0 + l15;      // V[k, n] = proj[k*PROJ + 128 + n]

  v8f acc = {};
#pragma unroll 4
  for (int k = 0; k < N; k += 4) {
    v2f a = *(const v2f*)(arow + k);
    v2f b;
    b.x = bcol[(size_t)(k + 2 * half) * PROJ];
    b.y = bcol[(size_t)(k + 2 * half + 1) * PROJ];
    acc = wmma_f32_16x16x4(a, b, acc);
  }
#pragma unroll
  for (int v = 0; v < 8; ++v) {
    const int m = m0 + v + 8 * half;
    out[(size_t)m * DV + n0 + l15] = acc[v];
  }
}

// -------------------------------------------------------------------------
// Launch. Workspace: proj (2 MB) + S (16 MB) = 18 MB.
// -------------------------------------------------------------------------
extern "C" void kernel_launch(void* const* d_in, const int* in_sizes, int n_in,
                              void* d_out, int out_size, void* d_ws, size_t ws_size,
                              hipStream_t stream) {
  const float* H = (const float*)d_in[0];   // (N, DIN)
  const float* D = (const float*)d_in[1];   // (N, N, DR)
  const float* W = (const float*)d_in[2];   // (PROJ, DIN)
  float* out = (float*)d_out;               // (N, DV)

  float* proj = (float*)d_ws;               // (N, PROJ)
  float* S    = proj + (size_t)N * PROJ;    // (N, N)

  proj_kernel<<<dim3(PROJ / 16, N / 16), 32, 0, stream>>>(H, W, proj);
  scores_kernel<<<dim3(N / 16, N / 16), 32, 0, stream>>>(proj, D, S);
  softmax_kernel<<<N, 256, 0, stream>>>(S);
  out_kernel<<<dim3(DV / 16, N / 16), 32, 0, stream>>>(S, proj, out);
}